// SC_Att_Bridge_70978629534275
// MI455X (gfx1250) — compile-verified
//
#include <hip/hip_runtime.h>
#include <cstdint>
#include <math.h>

typedef __attribute__((ext_vector_type(4))) float f4;
typedef __attribute__((ext_vector_type(2))) float v2f;
typedef __attribute__((ext_vector_type(8))) float v8f;
typedef __attribute__((ext_vector_type(4))) int   v4i;

typedef __attribute__((address_space(1))) v4i* gv4i_p;   // global
typedef __attribute__((address_space(3))) v4i* lv4i_p;   // LDS

// ---------------- CDNA5 helpers ----------------

__device__ inline void async_copy_f4(const float* g, float* lds_generic) {
#if defined(__has_builtin) && __has_builtin(__builtin_amdgcn_global_load_async_to_lds_b128)
  __builtin_amdgcn_global_load_async_to_lds_b128(
      (gv4i_p)(float*)g,            // global source (AS1), const dropped
      (lv4i_p)lds_generic,          // LDS destination (AS3)
      0, 0);
#else
  unsigned lo = (unsigned)(uintptr_t)lds_generic;  // low 32 bits = LDS byte offset
  asm volatile("global_load_async_to_lds_b128 %0, %1, off" :: "v"(lo), "v"(g) : "memory");
#endif
}

__device__ inline void wait_async0() {
#if defined(__has_builtin) && __has_builtin(__builtin_amdgcn_s_wait_asynccnt)
  __builtin_amdgcn_s_wait_asynccnt(0);
#else
  asm volatile("s_wait_asynccnt 0" ::: "memory");
#endif
}

__device__ inline f4 nt_load4(const float* p) {
  return __builtin_nontemporal_load((const f4*)p);
}
__device__ inline void nt_store4(float* p, f4 v) {
  __builtin_nontemporal_store(v, (f4*)p);
}
__device__ inline f4 vmax4(f4 a, f4 b) {
  f4 r;
  r.x = fmaxf(a.x, b.x); r.y = fmaxf(a.y, b.y);
  r.z = fmaxf(a.z, b.z); r.w = fmaxf(a.w, b.w);
  return r;
}

// ---------------- Kernel 1: channel mean + max -> att2[n][2][HW] ----------------
// grid.x = N * (HW/TPIX), block = 256. Each thread owns 4 pixels (f4).
// groups = 1024/TPIX channel-groups per block; LDS combine when groups > 1.
__global__ __launch_bounds__(256) void k_meanmax(const float* __restrict__ t,
                                                 float* __restrict__ att2,
                                                 int C, int HW, int TPIX) {
  const int tilesPerN = HW / TPIX;
  const int n  = blockIdx.x / tilesPerN;
  const int t0 = (blockIdx.x % tilesPerN) * TPIX;
  const int laneSz = TPIX >> 2;
  const int groups = 1024 / TPIX;
  const int tid = threadIdx.x;
  const int group = tid / laneSz;
  const int lane  = tid % laneSz;
  const int p = t0 + 4 * lane;

  const float* base = t + (size_t)n * C * HW + p;
  f4 s = {0.f, 0.f, 0.f, 0.f};
  f4 m = {-INFINITY, -INFINITY, -INFINITY, -INFINITY};
#pragma unroll 4
  for (int c = group; c < C; c += groups) {
    f4 v = *(const f4*)(base + (size_t)c * HW);
    s += v;
    m = vmax4(m, v);
  }

  __shared__ f4 rs[256];
  __shared__ f4 rm[256];
  if (groups > 1) {
    rs[tid] = s; rm[tid] = m;
    __syncthreads();
    if (group == 0) {
      for (int g = 1; g < groups; ++g) {
        s += rs[g * laneSz + lane];
        m = vmax4(m, rm[g * laneSz + lane]);
      }
    }
  }
  if (group == 0) {
    f4 avg = s * (1.0f / (float)C);
    *(f4*)&att2[(size_t)(2 * n) * HW + p]      = avg;
    *(f4*)&att2[(size_t)(2 * n) * HW + HW + p] = m;
  }
}

// ---------------- Kernel 2: dilated 7x7 conv + sigmoid via WMMA ----------------
// im2col GEMM: D(16x16) = A(16x4, duplicated weight rows) x B(4x16, patch cols),
// K = 98 taps padded to 100 -> 25 x v_wmma_f32_16x16x4_f32 per 16 output pixels.
// Per-tap (dh, dw, channel-offset) tables live in LDS so the hot loop has no
// div/mod chains; gathers are pipelined one WMMA step ahead.
__global__ __launch_bounds__(256) void k_conv_att(const float* __restrict__ att2,
                                                  const float* __restrict__ cw,
                                                  const float* __restrict__ cb,
                                                  float* __restrict__ att,
                                                  int H, int W) {
  __shared__ float sw[100];
  __shared__ int   s_dh[100];
  __shared__ int   s_dw[100];
  __shared__ int   s_coff[100];
  const int tid = threadIdx.x;
  if (tid < 100) {
    const int k = tid;
    sw[k] = (k < 98) ? cw[k] : 0.0f;
    if (k < 98) {
      const int ch = k / 49;
      const int r  = k % 49;
      s_dh[k]   = 3 * (r / 7) - 9;
      s_dw[k]   = 3 * (r % 7) - 9;
      s_coff[k] = ch * H * W;
    } else {
      s_dh[k] = 0x40000000;   // forces out-of-bounds -> contributes 0 (weight is 0 too)
      s_dw[k] = 0;
      s_coff[k] = 0;
    }
  }
  __syncthreads();

  const int HW = H * W;
  const int wave = tid >> 5;
  const int lane = tid & 31;
  const int gwave = blockIdx.x * 8 + wave;
  const int P = gwave * 16;            // 16 consecutive pixels, same row (W % 16 == 0)
  const int n   = P / HW;
  const int rem = P % HW;
  const int h  = rem / W;
  const int w0 = rem % W;
  const int col = lane & 15;
  const int hi  = lane >> 4;
  const int wp  = w0 + col;

  const float* xb = att2 + (size_t)(2 * n) * HW;

  auto gather = [&](int k) -> float {
    const int hh = h  + s_dh[k];
    const int ww = wp + s_dw[k];
    const bool ok = ((unsigned)hh < (unsigned)H) && ((unsigned)ww < (unsigned)W);
    const int hc = min(max(hh, 0), H - 1);
    const int wc = min(max(ww, 0), W - 1);
    const float v = xb[s_coff[k] + hc * W + wc];   // 32-bit index math
    return ok ? v : 0.0f;
  };

  v8f acc = {0.f, 0.f, 0.f, 0.f, 0.f, 0.f, 0.f, 0.f};
  const int kb = 2 * hi;               // lane's K-slot within each 4-wide chunk
  float bx = gather(kb);
  float by = gather(kb + 1);
#pragma unroll
  for (int kk = 0; kk < 25; ++kk) {
    v2f a; a.x = sw[4 * kk + kb]; a.y = sw[4 * kk + kb + 1];  // A[m][k] = w[k] (all rows)
    v2f b; b.x = bx; b.y = by;                                 // B[k][col]
    float nbx = 0.0f, nby = 0.0f;
    if (kk < 24) {                     // prefetch next chunk's gathers before the WMMA
      nbx = gather(4 * (kk + 1) + kb);
      nby = gather(4 * (kk + 1) + kb + 1);
    }
    acc = __builtin_amdgcn_wmma_f32_16x16x4_f32(
        /*neg_a=*/false, a, /*neg_b=*/false, b,
        /*c_mod=*/(short)0, acc, /*reuse_a=*/false, /*reuse_b=*/false);
    bx = nbx; by = nby;
  }
  // All D rows equal (A rows duplicated) -> every lane's acc[0] = y[pixel col]
  const float y = acc[0] + cb[0];
  const float s = 1.0f / (1.0f + __expf(-y));
  if (lane < 16) att[(size_t)n * HW + (size_t)h * W + w0 + col] = s;
}

// ---------------- Kernel 3: out = t * (1 + att), att staged via async->LDS ----------------
// grid.x = N * (HW/TPIX) * (C/cPB), block = 256. Non-temporal t loads / out stores.
__global__ __launch_bounds__(256) void k_gate(const float* __restrict__ t,
                                              const float* __restrict__ att,
                                              float* __restrict__ out,
                                              int C, int HW, int TPIX, int cPB) {
  const int tilesPerN = HW / TPIX;
  const int cBlocks = C / cPB;
  const int perN = tilesPerN * cBlocks;
  const int n = blockIdx.x / perN;
  const int r = blockIdx.x % perN;
  const int t0 = (r % tilesPerN) * TPIX;
  const int cb = (r / tilesPerN) * cPB;

  const int laneSz = TPIX >> 2;
  const int groups = 1024 / TPIX;
  const int tid = threadIdx.x;
  const int group = tid / laneSz;
  const int lane  = tid % laneSz;

  __shared__ float s_att[1024];
  if (tid < laneSz) {
    async_copy_f4(att + (size_t)n * HW + t0 + 4 * tid, &s_att[4 * tid]);
  }
  wait_async0();
  __syncthreads();

  f4 a = *(const f4*)&s_att[4 * lane];
  f4 gmul = a + 1.0f;   // gate + residual: t*sig + t = t*(1+sig)

  const float* tb = t   + (size_t)n * C * HW + t0 + 4 * lane;
  float*       ob = out + (size_t)n * C * HW + t0 + 4 * lane;
  for (int c = cb + group; c < cb + cPB; c += groups) {
    f4 v = nt_load4(tb + (size_t)c * HW);
    nt_store4(ob + (size_t)c * HW, v * gmul);
  }
}

// ---------------- Host ----------------

struct Shp { int C, H, W; };
static const Shp kShapes[5] = {
  {64, 256, 256}, {128, 128, 128}, {256, 64, 64}, {512, 32, 32}, {512, 16, 16}
};

extern "C" void kernel_launch(void* const* d_in, const int* in_sizes, int n_in,
                              void* d_out, int out_size, void* d_ws, size_t ws_size,
                              hipStream_t stream) {
  (void)in_sizes; (void)n_in; (void)out_size; (void)ws_size;
  const int N = 8;
  const float* cw = (const float*)d_in[5];   // [1,2,7,7] = 98 floats
  const float* cb = (const float*)d_in[6];   // [1]
  float* att2 = (float*)d_ws;                               // up to 8*2*65536*4 = 4 MB
  float* att  = (float*)((char*)d_ws + (size_t)(4u << 20)); // up to 2 MB

  size_t outOff = 0;
  for (int i = 0; i < 5; ++i) {
    const int C = kShapes[i].C, H = kShapes[i].H, W = kShapes[i].W;
    const int HW = H * W;
    const float* t = (const float*)d_in[i];
    float* o = (float*)d_out + outOff;

    const int TPIX = (HW < 1024) ? HW : 1024;
    const int tilesPerN = HW / TPIX;

    // 1) channel mean + max
    k_meanmax<<<dim3(N * tilesPerN), dim3(256), 0, stream>>>(t, att2, C, HW, TPIX);

    // 2) dilated conv + sigmoid (WMMA); 8 wave-tiles of 16 pixels per block
    k_conv_att<<<dim3((N * HW) / 128), dim3(256), 0, stream>>>(att2, cw, cb, att, H, W);

    // 3) gate + residual
    const int cPB = (TPIX < 1024) ? 16 : 8;   // keep cPB % (1024/TPIX) == 0
    k_gate<<<dim3(N * tilesPerN * (C / cPB)), dim3(256), 0, stream>>>(
        t, att, o, C, HW, TPIX, cPB);

    outOff += (size_t)N * C * HW;
  }
}